// OnlineTripletLoss_90082644066553
// MI455X (gfx1250) — compile-verified
//
#include <hip/hip_runtime.h>
#include <stdint.h>

typedef __attribute__((ext_vector_type(2))) float v2f;
typedef __attribute__((ext_vector_type(8))) float v8f;
typedef __attribute__((ext_vector_type(4))) unsigned int u32x4;
typedef __attribute__((ext_vector_type(8))) int i32x8;
typedef __attribute__((ext_vector_type(4))) int i32x4;
typedef unsigned long long u64;
typedef unsigned int u32;

#define BATCH 8192
#define DIMK  128
#define TRIP_MARGIN 1.0f

#define COLS_PER_WAVE 1024          // grid.y = 8 strips
#define STAGE_COLS    32            // cols per TDM stage (2 x 16-col WMMA tiles)
#define N_STAGES      (COLS_PER_WAVE / STAGE_COLS)
#define LDS_CSTRIDE   132           // 128 K-floats + 4 DWORD pad (bank-conflict-free)

// ---------------------------------------------------------------------------
// Helpers
// ---------------------------------------------------------------------------
__device__ __forceinline__ u64 shflx64(u64 v, int mask) {
    u32 lo = __shfl_xor((u32)(v & 0xFFFFFFFFull), mask, 32);
    u32 hi = __shfl_xor((u32)(v >> 32), mask, 32);
    return ((u64)hi << 32) | (u64)lo;
}

// TDM: DMA a [STAGE_COLS x 128] fp32 tile (row-major, 512B rows) into LDS,
// padding each 128-DWORD row with 4 DWORDs (=> LDS_CSTRIDE row pitch).
// Descriptor per CDNA5 ISA. clang-23 6-arg builtin:
// (u32x4 g0, i32x8 g1, i32x4 g2, i32x4 g3, i32x8 extra, i32 cpol).
__device__ __forceinline__ void tdm_load_tile(const float* gptr, u32 lds_byte_off) {
    const u64 ga = (u64)(uintptr_t)gptr;
    u32x4 g0;
    g0.x = 1u;                                            // count=1 (valid user D#)
    g0.y = lds_byte_off;                                  // lds_addr (bytes)
    g0.z = (u32)ga;                                       // global_addr[31:0]
    g0.w = (u32)((ga >> 32) & 0x1FFFFFFull) | (2u << 30); // addr[56:32] | type=2
    i32x8 g1;
    g1[0] = (2 << 16)        // data_size = 4B
          | (1 << 20)        // pad_enable
          | (6 << 22)        // pad_interval: every 128 DWORDs
          | (3 << 25);       // pad_amount: 4 DWORDs
    g1[1] = (int)(128u << 16);          // tensor_dim0 = 128 (low 16)
    g1[2] = (int)(8192u << 16);         // tensor_dim0 hi=0 | tensor_dim1 = 8192 (low 16)
    g1[3] = (int)(128u << 16);          // tensor_dim1 hi=0 | tile_dim0 = 128
    g1[4] = STAGE_COLS;                 // tile_dim1 = 32 | tile_dim2 = 0
    g1[5] = 128;                        // tensor_dim0_stride = 128 (low 32)
    g1[6] = 0;                          // stride hi | tensor_dim1_stride lo (unused, 2D)
    g1[7] = 0;
    const i32x4 gz4 = {0, 0, 0, 0};     // groups 2/3 unused for 2D tile
    const i32x8 gz8 = {0, 0, 0, 0, 0, 0, 0, 0};
    __builtin_amdgcn_tensor_load_to_lds(g0, g1, gz4, gz4, gz8, 0);
}

// ---------------------------------------------------------------------------
// Kernel 1: per-row squared norms + init mining workspace.
// ---------------------------------------------------------------------------
__launch_bounds__(256)
__global__ void tl_prep(const float* __restrict__ emb,
                        float* __restrict__ sq,
                        u64* __restrict__ pos,
                        u64* __restrict__ neg) {
    const int row  = (blockIdx.x * blockDim.x + threadIdx.x) >> 5;
    const int lane = threadIdx.x & 31;
    if (row >= BATCH) return;
    const float4 v = ((const float4*)(emb + (size_t)row * DIMK))[lane];
    float s = v.x * v.x + v.y * v.y + v.z * v.z + v.w * v.w;
    #pragma unroll
    for (int m = 16; m >= 1; m >>= 1) s += __shfl_xor(s, m, 32);
    if (lane == 0) {
        sq[row]  = s;
        pos[row] = 0ull;                      // hardest positive: running max
        neg[row] = 0xFFFFFFFFFFFFFFFFull;     // hardest negative: running min
    }
}

// ---------------------------------------------------------------------------
// Kernel 2: fused fp32 WMMA Gram-matrix + batch-hard mining.
// Grid (64, 8), block 256 = 8 waves.
//   wave w: rows r0 = blockIdx.x*128 + w*16 ; cols = blockIdx.y*1024 strip.
// B operand is TDM-DMA'd into double-buffered LDS (shared by all 8 waves),
// A strip (16x128) lives in VGPRs, K=128 = 32 chained V_WMMA_F32_16X16X4_F32.
// Mining on clamped d^2 packed as (f32bits<<32)|col for u64 max/min atomics.
// ---------------------------------------------------------------------------
__launch_bounds__(256)
__global__ void tl_mine(const float* __restrict__ emb,
                        const long long* __restrict__ tgt,
                        const float* __restrict__ sq,
                        u64* __restrict__ pos,
                        u64* __restrict__ neg) {
    __shared__ __align__(16) float ldsB[2][STAGE_COLS * LDS_CSTRIDE];
    __shared__ float ldsSq[COLS_PER_WAVE];
    __shared__ int   ldsTg[COLS_PER_WAVE];

    const int lane = threadIdx.x & 31;
    const int wv   = threadIdx.x >> 5;
    const int m16  = lane & 15;   // fragment row/col index
    const int h    = lane >> 4;   // half-wave
    const int r0   = blockIdx.x * 128 + wv * 16;
    const int c0   = blockIdx.y * COLS_PER_WAVE;

    // ---- stage column-side sq/target into LDS (once per WG) ----
    for (int i = threadIdx.x; i < COLS_PER_WAVE; i += 256) {
        ldsSq[i] = sq[c0 + i];
        ldsTg[i] = (int)tgt[c0 + i];
    }

    // ---- kick off TDM DMA of stage 0 ----
    if (wv == 0)
        tdm_load_tile(emb + (size_t)c0 * DIMK, (u32)(uintptr_t)&ldsB[0][0]);

    // ---- preload A fragments: this wave's 16x128 row strip ----
    // A(16x4 fp32): lane m = lane&15; vgpr j holds K = 4*ks + 2*h + j.
    v2f a[32];
    const size_t abase = (size_t)(r0 + m16) * DIMK + 2 * h;
    #pragma unroll
    for (int ks = 0; ks < 32; ++ks)
        a[ks] = *(const v2f*)(emb + abase + ks * 4);

    // per-row scalars: row of acc element (v, half h) is r0 + v + 8*h
    float srow[8]; int trow[8];
    #pragma unroll
    for (int v = 0; v < 8; ++v) {
        const int r = r0 + v + 8 * h;
        srow[v] = sq[r];
        trow[v] = (int)tgt[r];
    }

    u64 pbest[8], nbest[8];
    #pragma unroll
    for (int v = 0; v < 8; ++v) { pbest[v] = 0ull; nbest[v] = ~0ull; }

    for (int s = 0; s < N_STAGES; ++s) {
        const int buf = s & 1;
        if (wv == 0) __builtin_amdgcn_s_wait_tensorcnt(0);  // stage s DMA done
        __syncthreads();   // buf[s] ready for all; buf[s+1] free (read last stage)
        if (wv == 0 && s + 1 < N_STAGES)
            tdm_load_tile(emb + (size_t)(c0 + (s + 1) * STAGE_COLS) * DIMK,
                          (u32)(uintptr_t)&ldsB[(s + 1) & 1][0]);

        #pragma unroll
        for (int nt = 0; nt < 2; ++nt) {
            // B(4x16 fp32) fragments from LDS: col n = lane&15, K = 4*ks+2h+j
            const float* bbase =
                &ldsB[buf][(nt * 16 + m16) * LDS_CSTRIDE + 2 * h];
            v2f b[32];
            #pragma unroll
            for (int ks = 0; ks < 32; ++ks)
                b[ks] = *(const v2f*)(bbase + ks * 4);

            v8f acc = {};
            #pragma unroll
            for (int ks = 0; ks < 32; ++ks)
                acc = __builtin_amdgcn_wmma_f32_16x16x4_f32(
                    false, a[ks], false, b[ks], (short)0, acc, false, false);

            const int  lc   = s * STAGE_COLS + nt * 16 + m16; // col within strip
            const int  bcol = c0 + lc;                        // global column
            const float scol = ldsSq[lc];
            const int   tcol = ldsTg[lc];

            #pragma unroll
            for (int v = 0; v < 8; ++v) {
                const int i = r0 + v + 8 * h;                 // row of acc[v]
                float d2 = srow[v] + scol - 2.0f * acc[v];
                d2 = fmaxf(d2, 0.0f);
                const u64 packed = ((u64)__float_as_uint(d2) << 32) | (u32)bcol;
                const bool same = (trow[v] == tcol);
                const u64 pc = (same && (i != bcol)) ? packed : 0ull;
                const u64 nc = (!same) ? packed : ~0ull;
                pbest[v] = pbest[v] > pc ? pbest[v] : pc;
                nbest[v] = nbest[v] < nc ? nbest[v] : nc;
            }
        }
    }

    // reduce across the 16 lanes of each half-wave, then combine strips globally
    #pragma unroll
    for (int v = 0; v < 8; ++v) {
        u64 p = pbest[v], n = nbest[v];
        #pragma unroll
        for (int m = 8; m >= 1; m >>= 1) {
            const u64 po = shflx64(p, m);
            const u64 no = shflx64(n, m);
            p = p > po ? p : po;
            n = n < no ? n : no;
        }
        if (m16 == 0) {
            const int i = r0 + v + 8 * h;
            atomicMax((unsigned long long*)&pos[i], (unsigned long long)p);
            atomicMin((unsigned long long*)&neg[i], (unsigned long long)n);
        }
    }
}

// ---------------------------------------------------------------------------
// Kernel 3: loss = mean(relu(sqrt(d2_pos) - sqrt(d2_neg) + margin)).
// ---------------------------------------------------------------------------
__launch_bounds__(256)
__global__ void tl_loss(const u64* __restrict__ pos,
                        const u64* __restrict__ neg,
                        float* __restrict__ out) {
    __shared__ float red[256];
    float s = 0.0f;
    for (int i = threadIdx.x; i < BATCH; i += 256) {
        const float dp = sqrtf(__uint_as_float((u32)(pos[i] >> 32)));
        const float dn = sqrtf(__uint_as_float((u32)(neg[i] >> 32)));
        s += fmaxf(dp - dn + TRIP_MARGIN, 0.0f);
    }
    red[threadIdx.x] = s;
    __syncthreads();
    #pragma unroll
    for (int st = 128; st > 0; st >>= 1) {
        if ((int)threadIdx.x < st) red[threadIdx.x] += red[threadIdx.x + st];
        __syncthreads();
    }
    if (threadIdx.x == 0) out[0] = red[0] * (1.0f / (float)BATCH);
}

// ---------------------------------------------------------------------------
extern "C" void kernel_launch(void* const* d_in, const int* in_sizes, int n_in,
                              void* d_out, int out_size, void* d_ws, size_t ws_size,
                              hipStream_t stream) {
    const float*     emb = (const float*)d_in[0];
    const long long* tgt = (const long long*)d_in[1];

    u64*   pos = (u64*)d_ws;            // [8192] packed (d2, argmax col)
    u64*   neg = pos + BATCH;           // [8192] packed (d2, argmin col)
    float* sq  = (float*)(neg + BATCH); // [8192] row squared norms

    tl_prep<<<dim3(BATCH / 8), 256, 0, stream>>>(emb, sq, pos, neg);
    tl_mine<<<dim3(64, 8), 256, 0, stream>>>(emb, tgt, sq, pos, neg);
    tl_loss<<<dim3(1), 256, 0, stream>>>(pos, neg, (float*)d_out);
}